// DCFeasibilityLayer_26448408609094
// MI455X (gfx1250) — compile-verified
//
#include <hip/hip_runtime.h>

typedef float v2f __attribute__((ext_vector_type(2)));
typedef float v8f __attribute__((ext_vector_type(8)));

#ifndef N_ITER
#define N_ITER 12
#endif

// One block = 64 threads = 2 wave32 waves; block handles 16 batch rows.
// Wave w owns columns [128w, 128w+128); lane l owns row (l&15) and 64
// contiguous columns. p / p_pred live in VGPRs across all 12 iterations.
// Row sums are done on the matrix pipe: V_WMMA_F32_16X16X4_F32 with A=ones,
// so each lane's D[...] element 0 is exactly its row's partial sum (fp32).
__global__ __launch_bounds__(64) void dc_feasibility_kernel(
    const float* __restrict__ p_pred,
    const float* __restrict__ total_load,
    const float* __restrict__ p_min,
    const float* __restrict__ p_max,
    float* __restrict__ out)
{
  constexpr int   NG   = 256;
  constexpr float LRP  = 0.5f;
  constexpr float LRD  = 0.8f;
  constexpr float LCLP = 1000000.0f;
  constexpr float EPS  = 1e-8f;

  __shared__ v2f   s_lo[NG / 2];
  __shared__ v2f   s_hi[NG / 2];
  __shared__ float s_part[2][2][16];   // [buf][half][row]

  const int tid  = threadIdx.x;        // 0..63
  const int wave = tid >> 5;           // column half: 0 or 1
  const int lane = tid & 31;
  const int row  = lane & 15;          // batch row within this group
  const int colb = wave * 128 + (lane >> 4) * 64;  // lane's first column

  // Stage clip bounds into LDS (per-iteration reads are lane-broadcasts).
  {
    const v2f* gl = reinterpret_cast<const v2f*>(p_min);
    const v2f* gh = reinterpret_cast<const v2f*>(p_max);
    #pragma unroll
    for (int i = 0; i < 2; ++i) {      // 64 threads * 2 = 128 v2f = 256 floats
      s_lo[tid + 64 * i] = gl[tid + 64 * i];
      s_hi[tid + 64 * i] = gh[tid + 64 * i];
    }
  }
  __syncthreads();

  const size_t grow = (size_t)blockIdx.x * 16 + (size_t)row;
  const float4* src = reinterpret_cast<const float4*>(p_pred + grow * NG + colb);

  v2f p[32];    // current iterate: 64 floats, cols [colb, colb+64)
  v2f pr[32];   // p_pred (kept resident: needed every iteration)
  #pragma unroll
  for (int i = 0; i < 16; ++i) {
    float4 v = src[i];
    pr[2 * i + 0] = v2f{v.x, v.y};
    pr[2 * i + 1] = v2f{v.z, v.w};
    p[2 * i + 0]  = pr[2 * i + 0];
    p[2 * i + 1]  = pr[2 * i + 1];
  }

  const float loadv = total_load[grow];
  const v2f*  lo    = s_lo + (colb >> 1);
  const v2f*  hi    = s_hi + (colb >> 1);
  const v2f   onesA = {1.0f, 1.0f};

  float lam = 0.0f;
  float res = 0.0f;

  #pragma unroll 1
  for (int it = 0; it < N_ITER; ++it) {
    v8f a0 = {};
    v8f a1 = {};
    #pragma unroll
    for (int j = 0; j < 32; j += 2) {
      {
        v2f l = lo[j], h = hi[j];
        v2f g, t;
        g.x = (p[j].x - pr[j].x) + lam;
        g.y = (p[j].y - pr[j].y) + lam;
        t.x = fminf(fmaxf(p[j].x - LRP * g.x, l.x), h.x);
        t.y = fminf(fmaxf(p[j].y - LRP * g.y, l.y), h.y);
        p[j] = t;
        a0 = __builtin_amdgcn_wmma_f32_16x16x4_f32(false, onesA, false, t,
                                                   (short)0, a0, false, false);
      }
      {
        v2f l = lo[j + 1], h = hi[j + 1];
        v2f g, t;
        g.x = (p[j + 1].x - pr[j + 1].x) + lam;
        g.y = (p[j + 1].y - pr[j + 1].y) + lam;
        t.x = fminf(fmaxf(p[j + 1].x - LRP * g.x, l.x), h.x);
        t.y = fminf(fmaxf(p[j + 1].y - LRP * g.y, l.y), h.y);
        p[j + 1] = t;
        a1 = __builtin_amdgcn_wmma_f32_16x16x4_f32(false, onesA, false, t,
                                                   (short)0, a1, false, false);
      }
    }
    // Each lane's element 0 of D is its row's partial sum over this wave's
    // 128 columns (all D rows are identical since A is all-ones).
    const float halfsum = a0[0] + a1[0];
    s_part[it & 1][wave][row] = halfsum;       // lanes r and r+16 write the
    __syncthreads();                           // same value: benign
    const float tot = halfsum + s_part[it & 1][wave ^ 1][row];
    res = tot - loadv;
    lam = fminf(fmaxf(lam + LRD * res, -LCLP), LCLP);
  }

  // Final slack correction. res from the last iteration equals the
  // reference's post-loop res (p unchanged in between).
  v2f m[32];
  v8f sa = {};
  #pragma unroll
  for (int j = 0; j < 32; ++j) {
    v2f l = lo[j], h = hi[j];
    v2f mm;
    mm.x = (p[j].x > l.x + EPS && p[j].x < h.x - EPS) ? 1.0f : 0.0f;
    mm.y = (p[j].y > l.y + EPS && p[j].y < h.y - EPS) ? 1.0f : 0.0f;
    m[j] = mm;
    sa = __builtin_amdgcn_wmma_f32_16x16x4_f32(false, onesA, false, mm,
                                               (short)0, sa, false, false);
  }
  const float shalf = sa[0];
  s_part[0][wave][row] = shalf;   // last read of s_part[0] was it=10: safe
  __syncthreads();
  float ssum = shalf + s_part[0][wave ^ 1][row];
  ssum = fmaxf(ssum, 1.0f);       // == where(ss==0, 1, ss) for ss >= 0
  const float corr = res / ssum;

  float4* dst = reinterpret_cast<float4*>(out + grow * NG + colb);
  #pragma unroll
  for (int i = 0; i < 16; ++i) {
    float4 v;
    v.x = p[2 * i + 0].x - corr * m[2 * i + 0].x;
    v.y = p[2 * i + 0].y - corr * m[2 * i + 0].y;
    v.z = p[2 * i + 1].x - corr * m[2 * i + 1].x;
    v.w = p[2 * i + 1].y - corr * m[2 * i + 1].y;
    dst[i] = v;
  }
}

extern "C" void kernel_launch(void* const* d_in, const int* in_sizes, int n_in,
                              void* d_out, int out_size, void* d_ws, size_t ws_size,
                              hipStream_t stream) {
  (void)n_in; (void)out_size; (void)d_ws; (void)ws_size;
  const float* p_pred     = (const float*)d_in[0];
  const float* total_load = (const float*)d_in[1];
  const float* p_min      = (const float*)d_in[2];
  const float* p_max      = (const float*)d_in[3];
  float* out = (float*)d_out;

  const int B = in_sizes[1];           // 65536 batch rows (divisible by 16)
  dim3 grid(B / 16), block(64);
  hipLaunchKernelGGL(dc_feasibility_kernel, grid, block, 0, stream,
                     p_pred, total_load, p_min, p_max, out);
}